// SpatialQFM_43087111913716
// MI455X (gfx1250) — compile-verified
//
#include <hip/hip_runtime.h>
#include <hip/hip_bf16.h>

// ---------------------------------------------------------------------------
// Compile-time Heisenberg back-propagation of the 40 Z-string observables
// through the 48-CNOT entangler (2 layers of HPAIRS then VPAIRS on 4x4 grid).
// Rule: conjugating Z-string by CNOT(c,t): if mask has bit t, toggle bit c.
// Gates are traversed in reverse application order.
// ---------------------------------------------------------------------------
struct Masks { unsigned m[40]; };

constexpr Masks make_masks() {
    int gc[48] = {}, gt[48] = {};
    int idx = 0;
    for (int layer = 0; layer < 2; ++layer) {
        for (int r = 0; r < 4; ++r)
            for (int c = 0; c < 3; ++c) { gc[idx] = r * 4 + c; gt[idx] = r * 4 + c + 1; ++idx; }
        for (int r = 0; r < 3; ++r)
            for (int c = 0; c < 4; ++c) { gc[idx] = r * 4 + c; gt[idx] = (r + 1) * 4 + c; ++idx; }
    }
    unsigned init[40] = {};
    for (int q = 0; q < 16; ++q) init[q] = 1u << q;
    int f = 16;
    for (int r = 0; r < 4; ++r)
        for (int c = 0; c < 3; ++c) { init[f++] = (1u << (r * 4 + c)) | (1u << (r * 4 + c + 1)); }
    for (int r = 0; r < 3; ++r)
        for (int c = 0; c < 4; ++c) { init[f++] = (1u << (r * 4 + c)) | (1u << ((r + 1) * 4 + c)); }
    Masks M = {};
    for (int i = 0; i < 40; ++i) {
        unsigned mask = init[i];
        for (int g = 47; g >= 0; --g)
            if ((mask >> gt[g]) & 1u) mask ^= (1u << gc[g]);
        M.m[i] = mask;
    }
    return M;
}

__constant__ Masks MASKS = make_masks();

// ---------------------------------------------------------------------------
// Kernel 1: per-(circuit,qubit) <Z> table.
// theta = (gray+1)*pi/2  =>  cos(theta) = -sin(gray*pi/2).
// cosz layout: [512 circuits][16 qubits], circuit n = b*64 + (py*8+px).
// ---------------------------------------------------------------------------
__global__ void cosz_kernel(const float* __restrict__ xin, float* __restrict__ cosz) {
    const int idx = blockIdx.x * blockDim.x + threadIdx.x;  // 0..8191
    if (idx >= 8 * 64 * 16) return;
    const int q = idx & 15;
    const int n = idx >> 4;
    const int b = n >> 6;
    const int p = n & 63;
    const int py = p >> 3, px = p & 7;
    const int iy = q >> 2, ix = q & 3;
    const int y = py * 4 + iy, xx = px * 4 + ix;
    const int base = ((b * 3) * 32 + y) * 32 + xx;          // channel stride = 1024
    const float g = (xin[base] + xin[base + 1024] + xin[base + 2048]) * (1.0f / 3.0f);
    cosz[idx] = -__sinf(g * 1.57079632679489662f);
}

// ---------------------------------------------------------------------------
// Kernel 2: feature = product of cos(theta_q) over the observable's Z-mask,
// fused with the 4-patch average pooling. Output vpad: [16][640] with rows
// 8..15 zeroed so the WMMA A-operand needs no lane predication.
// ---------------------------------------------------------------------------
__global__ void feat_kernel(const float* __restrict__ cosz, float* __restrict__ vpad) {
    const int idx = blockIdx.x * blockDim.x + threadIdx.x;  // 0..10239
    if (idx >= 16 * 640) return;
    if (idx >= 8 * 640) { vpad[idx] = 0.0f; return; }       // zero pad rows 8..15
    const int b = idx / 640;
    const int r = idx - b * 640;
    const int g = r / 40;
    const int f = r - g * 40;
    const unsigned mask = MASKS.m[f];
    float acc = 0.0f;
    #pragma unroll
    for (int j = 0; j < 4; ++j) {
        const float* cz = cosz + ((b * 64 + 4 * g + j) << 4);
        float prod = 1.0f;
        unsigned mm = mask;
        while (mm) {
            const int q = __ffs((int)mm) - 1;
            prod *= cz[q];
            mm &= mm - 1;
        }
        acc += prod;
    }
    vpad[idx] = 0.25f * acc;
}

// ---------------------------------------------------------------------------
// Kernel 3: MLP.  Layer 1 = vpad(16x640) @ W1^T(640x64) via
// V_WMMA_F32_16X16X4_F32. One block, 4 wave32s; wave w owns columns
// [16w, 16w+16). A 16x4 f32 operand: lane&15 = M row, lane>>4 selects K pair
// {0,1}|{2,3}; B 4x16 mirrors with lane&15 = N col. D 16x16 f32: VGPR r holds
// row r + 8*(lane>>4), col = lane&15. All loads unconditional 8-byte B64s
// (A is pre-zero-padded), K-loop unrolled 4x with loads batched ahead of the
// WMMA chain.
// ---------------------------------------------------------------------------
typedef __attribute__((ext_vector_type(2))) float v2f;
typedef __attribute__((ext_vector_type(8))) float v8f;

__global__ __launch_bounds__(128) void mlp_kernel(const float* __restrict__ vpad,
                                                  const float* __restrict__ W1,
                                                  const float* __restrict__ b1,
                                                  const float* __restrict__ W2,
                                                  const float* __restrict__ b2,
                                                  float* __restrict__ out) {
    __shared__ float hsh[8][64];
    const int tid   = threadIdx.x;
    const int wave  = tid >> 5;
    const int lane  = tid & 31;
    const int m     = lane & 15;        // A row / B col within tile
    const int khalf = lane >> 4;        // 0 or 1
    const int col   = wave * 16 + m;    // hidden unit 0..63

    const float* arow = vpad + m * 640 + 2 * khalf;   // even float offset -> 8B aligned
    const float* brow = W1 + col * 640 + 2 * khalf;

    v8f acc = {};
    for (int kk = 0; kk < 640; kk += 16) {
        v2f a[4], bm[4];
        #pragma unroll
        for (int u = 0; u < 4; ++u) {
            a[u]  = *(const v2f*)(arow + kk + 4 * u);
            bm[u] = *(const v2f*)(brow + kk + 4 * u);
        }
        #pragma unroll
        for (int u = 0; u < 4; ++u) {
            acc = __builtin_amdgcn_wmma_f32_16x16x4_f32(
                /*neg_a=*/false, a[u], /*neg_b=*/false, bm[u],
                /*c_mod=*/(short)0, acc, /*reuse_a=*/false, /*reuse_b=*/false);
        }
    }

    #pragma unroll
    for (int r = 0; r < 8; ++r) {
        const int row = r + 8 * khalf;  // D-matrix row for this lane/VGPR
        if (row < 8) {
            hsh[row][col] = fmaxf(acc[r] + b1[col], 0.0f);
        }
    }
    __syncthreads();

    // Layer 2: (8x64) @ W2^T(64x10) + b2 -> (8x10), 80 scalar dot products.
    if (tid < 80) {
        const int b = tid / 10, o = tid - b * 10;
        float s = b2[o];
        #pragma unroll
        for (int k = 0; k < 64; ++k) s = fmaf(hsh[b][k], W2[o * 64 + k], s);
        out[b * 10 + o] = s;
    }
}

// ---------------------------------------------------------------------------
// Launch: inputs are {x, W1, b1, W2, b2} (all float32), output (8,10) float32.
// Workspace: cosz (8192 f32 = 32KB) then vpad (10240 f32 = 40KB).
// ---------------------------------------------------------------------------
extern "C" void kernel_launch(void* const* d_in, const int* in_sizes, int n_in,
                              void* d_out, int out_size, void* d_ws, size_t ws_size,
                              hipStream_t stream) {
    const float* x  = (const float*)d_in[0];
    const float* W1 = (const float*)d_in[1];
    const float* b1 = (const float*)d_in[2];
    const float* W2 = (const float*)d_in[3];
    const float* b2 = (const float*)d_in[4];
    float* outp = (float*)d_out;

    float* cosz = (float*)d_ws;     // 8*64*16 = 8192 floats
    float* vpad = cosz + 8192;      // 16*640  = 10240 floats (rows 8..15 zero)

    cosz_kernel<<<dim3(32), dim3(256), 0, stream>>>(x, cosz);
    feat_kernel<<<dim3(40), dim3(256), 0, stream>>>(cosz, vpad);
    mlp_kernel<<<dim3(1), dim3(128), 0, stream>>>(vpad, W1, b1, W2, b2, outp);
}